// NonlinHRULayer_65137474011629
// MI455X (gfx1250) — compile-verified
//
#include <hip/hip_runtime.h>
#include <math.h>

// Problem dims (fixed by the reference)
#define LL 8192
#define HH 512
#define PP 1024

typedef __attribute__((ext_vector_type(2))) float v2f;
typedef __attribute__((ext_vector_type(8))) float v8f;

// ---------------------------------------------------------------------------
// Kernel 1: Bu[l,p] = sum_h u[l,h] * B[p,h]
// One wave computes one 16x16 tile of Bu using V_WMMA_F32_16X16X4_F32.
// A = u tile (M=l, K=h), B = Bmat tile as Bᵀ (K=h, N=p).
// Both operands are read as contiguous float2 along h (K).
// ---------------------------------------------------------------------------
__global__ __launch_bounds__(128) void gemm_bu_kernel(
    const float* __restrict__ u,      // (L,H)
    const float* __restrict__ Bmat,   // (P,H)
    float* __restrict__ Bu)           // (L,P)
{
    const int wave_in_block = threadIdx.x >> 5;           // 4 waves / block
    const int tile_id = blockIdx.x * 4 + wave_in_block;   // 0 .. 512*64-1
    const int tileM = tile_id / (PP / 16);                // l-tile
    const int tileN = tile_id % (PP / 16);                // p-tile
    const int l0 = tileM * 16;
    const int p0 = tileN * 16;

    const int lane = threadIdx.x & 31;
    const int half = lane >> 4;      // 0: K={0,1}, 1: K={2,3}
    const int r    = lane & 15;      // M (for A) / N (for B)

    const float* aRow = u    + (size_t)(l0 + r) * HH + 2 * half;
    const float* bRow = Bmat + (size_t)(p0 + r) * HH + 2 * half;

    v8f acc = {};
    #pragma unroll 4
    for (int h = 0; h < HH; h += 4) {
        v2f a = *(const v2f*)(aRow + h);
        v2f b = *(const v2f*)(bRow + h);
        acc = __builtin_amdgcn_wmma_f32_16x16x4_f32(
            false, a, false, b, (short)0, acc, false, false);
    }

    // D layout: VGPR i -> row (i + 8*half), col r
    const int col = p0 + r;
    #pragma unroll
    for (int i = 0; i < 8; ++i) {
        const int row = l0 + i + 8 * half;
        Bu[(size_t)row * PP + col] = acc[i];
    }
}

// ---------------------------------------------------------------------------
// Kernel 2: leapfrog scan. Each thread owns one channel p (independent
// recurrence), loops sequentially over L. Bu loads are coalesced across
// threads; prefetch keeps the stream ahead of the tanh dependency chain.
// ---------------------------------------------------------------------------
__global__ __launch_bounds__(256) void leapfrog_scan_kernel(
    const float* __restrict__ Bu,     // (L,P)
    const float* __restrict__ W,      // (P)
    const float* __restrict__ bvec,   // (P)
    const float* __restrict__ cvec,   // (P)
    const float* __restrict__ alpha,  // (1)
    const float* __restrict__ step,   // (1)
    float* __restrict__ ys)           // (L,P)
{
    const int p = blockIdx.x * 256 + threadIdx.x;   // 4 blocks * 256 = P
    const float al = alpha[0];
    const float st = step[0];
    const float hst = 0.5f * st;
    const float w = W[p];
    const float bb = bvec[p];
    // c_eff = sigmoid(sigmoid(c))
    const float s1 = 1.0f / (1.0f + __expf(-cvec[p]));
    const float ce = 1.0f / (1.0f + __expf(-s1));

    float z = 0.0f, y = 0.0f;
    for (int t = 0; t < LL; ++t) {
        if (t + 16 < LL)
            __builtin_prefetch(&Bu[(size_t)(t + 16) * PP + p], 0, 0);
        const float bu = Bu[(size_t)t * PP + p];
        const float y_half = fmaf(hst, z, y);
        const float grad = fmaf(al, y, ce * tanhf(fmaf(w, y, bu + bb)));
        z = fmaf(-st, grad, z);
        y = fmaf(hst, z, y_half);
        ys[(size_t)t * PP + p] = y;
    }
}

// ---------------------------------------------------------------------------
// Kernel 3: out[l,h] = sum_p ys[l,p] * C[h,p] + D[h]*u[l,h]
// A = ys tile (M=l, K=p), B = Cᵀ tile (K=p, N=h); C row-major (H,P) means
// Cᵀ(k,n) = C[(h0+n)*P + p0+k] -> contiguous float2 along K for each lane.
// ---------------------------------------------------------------------------
__global__ __launch_bounds__(128) void gemm_out_kernel(
    const float* __restrict__ ys,     // (L,P)
    const float* __restrict__ Cmat,   // (H,P)
    const float* __restrict__ Dvec,   // (H)
    const float* __restrict__ u,      // (L,H)
    float* __restrict__ out)          // (L,H)
{
    const int wave_in_block = threadIdx.x >> 5;
    const int tile_id = blockIdx.x * 4 + wave_in_block;   // 0 .. 512*32-1
    const int tileM = tile_id / (HH / 16);                // l-tile
    const int tileN = tile_id % (HH / 16);                // h-tile
    const int l0 = tileM * 16;
    const int h0 = tileN * 16;

    const int lane = threadIdx.x & 31;
    const int half = lane >> 4;
    const int r    = lane & 15;

    const float* aRow = ys   + (size_t)(l0 + r) * PP + 2 * half;
    const float* bRow = Cmat + (size_t)(h0 + r) * PP + 2 * half;

    v8f acc = {};
    #pragma unroll 4
    for (int p = 0; p < PP; p += 4) {
        v2f a = *(const v2f*)(aRow + p);
        v2f b = *(const v2f*)(bRow + p);
        acc = __builtin_amdgcn_wmma_f32_16x16x4_f32(
            false, a, false, b, (short)0, acc, false, false);
    }

    const int col = h0 + r;
    const float d = Dvec[col];
    #pragma unroll
    for (int i = 0; i < 8; ++i) {
        const int row = l0 + i + 8 * half;
        out[(size_t)row * HH + col] =
            acc[i] + d * u[(size_t)row * HH + col];
    }
}

// ---------------------------------------------------------------------------
extern "C" void kernel_launch(void* const* d_in, const int* in_sizes, int n_in,
                              void* d_out, int out_size, void* d_ws, size_t ws_size,
                              hipStream_t stream) {
    const float* u     = (const float*)d_in[0];  // (L,H)
    const float* Bmat  = (const float*)d_in[1];  // (P,H)
    const float* Cmat  = (const float*)d_in[2];  // (H,P)
    const float* Dvec  = (const float*)d_in[3];  // (H)
    const float* W     = (const float*)d_in[4];  // (P)
    const float* bvec  = (const float*)d_in[5];  // (P)
    const float* cvec  = (const float*)d_in[6];  // (P)
    const float* alpha = (const float*)d_in[7];  // (1)
    const float* step  = (const float*)d_in[8];  // (1)
    float* out = (float*)d_out;

    float* Bu = (float*)d_ws;                    // L*P floats = 32 MB
    float* ys = Bu + (size_t)LL * PP;            // L*P floats = 32 MB

    // Bu = u @ B^T : 512 x 64 tiles, 4 waves/block of 128 threads
    gemm_bu_kernel<<<(512 * 64) / 4, 128, 0, stream>>>(u, Bmat, Bu);

    // Sequential leapfrog over L, parallel over P channels
    leapfrog_scan_kernel<<<PP / 256, 256, 0, stream>>>(Bu, W, bvec, cvec,
                                                       alpha, step, ys);

    // out = ys @ C^T + D*u : 512 x 32 tiles
    gemm_out_kernel<<<(512 * 32) / 4, 128, 0, stream>>>(ys, Cmat, Dvec, u, out);
}